// Classifier0_1443109012173
// MI455X (gfx1250) — compile-verified
//
#include <hip/hip_runtime.h>
#include <math.h>

// Problem constants (match reference)
#define NB    256      // batch
#define SS    512      // spatial side
#define HH    256      // half side
#define NOUT  4
#define COUT  4
#define NCLS  10

typedef __attribute__((ext_vector_type(4))) float f4;
typedef __attribute__((ext_vector_type(2))) float v2f;
typedef __attribute__((ext_vector_type(8))) float v8f;

// ---------------------------------------------------------------------------
// Stage 1: quadrant sums. Memory-bound streaming reduction.
//   grid = (4 quadrants, 256 samples), block = 512 threads (16 wave32s).
//   Each block reduces one 256x256 quadrant (65536 floats = 16384 float4).
//   64 float4-lanes cover one quadrant row, 8 rows in flight per iteration,
//   32 iterations -> fully coalesced global_load_b128, non-temporal (zero
//   reuse; keep the 192MB L2 clean). Deterministic fixed-shape reduction,
//   single writer per agg cell (no float atomics).
// ---------------------------------------------------------------------------
__global__ __launch_bounds__(512) void quad_sum_kernel(
    const float* __restrict__ x, float* __restrict__ agg) {
  const int q   = blockIdx.x;   // 0=TL, 1=BL, 2=BR, 3=TR (reference order)
  const int n   = blockIdx.y;
  const int tid = threadIdx.x;

  const int r0  = (q == 1 || q == 2) ? HH : 0;       // row block
  const int c04 = (q == 2 || q == 3) ? (HH / 4) : 0; // col block in float4 units

  const f4* __restrict__ x4 = reinterpret_cast<const f4*>(x);
  const size_t sample_base = (size_t)n * (SS * SS / 4);

  const int rlane = tid >> 6;   // 0..7  (row within 8-row group)
  const int clane = tid & 63;   // 0..63 (float4 column within quadrant row)

  f4 acc;
  acc[0] = 0.f; acc[1] = 0.f; acc[2] = 0.f; acc[3] = 0.f;

#pragma unroll 8
  for (int it = 0; it < 32; ++it) {
    const int row = r0 + it * 8 + rlane;
    const f4* p = x4 + (sample_base + (size_t)row * (SS / 4) + c04 + clane);
    f4 v = __builtin_nontemporal_load(p);
    acc[0] += v[0]; acc[1] += v[1]; acc[2] += v[2]; acc[3] += v[3];
  }
  float s = (acc[0] + acc[1]) + (acc[2] + acc[3]);

  __shared__ float red[512];
  red[tid] = s;
  __syncthreads();
#pragma unroll
  for (int off = 256; off >= 32; off >>= 1) {
    if (tid < off) red[tid] += red[tid + off];
    __syncthreads();
  }
  if (tid < 32) {
    float v = red[tid];
#pragma unroll
    for (int o = 16; o > 0; o >>= 1) v += __shfl_down(v, o, 32);
    if (tid == 0) agg[n * NOUT + q] = v;  // single deterministic writer
  }
}

// ---------------------------------------------------------------------------
// Stage 2: weight-norm affine + final FC via V_WMMA_F32_16X16X4_F32.
//   out[256,10] = (agg*w + b_fgl)[256,16] @ W_fc^T[16,10] + b_fc
//   One block, 16 waves; wave w owns M-tile rows [16w, 16w+16).
//   K=16 as 4 chained K=4 WMMAs; bias folded into C; cols 10..15 padded with
//   zero B columns. fp32 end to end. CIN==1 so g*v/||v|| == g*sign(v):
//   use copysignf instead of the IEEE sqrt/div expansion. Masked (not
//   branchy) handling of the NCLS<16 padding keeps EXEC full and the code
//   straight-line around the WMMAs.
//
//   A 16x4 f32 layout: lanes 0-15 -> (M=lane, K=0/1 in vgpr0/1),
//                      lanes 16-31 -> (M=lane-16, K=2/3).
//   B 4x16 / C,D 16x16: rows striped across lanes; N = lane & 15.
// ---------------------------------------------------------------------------
__global__ __launch_bounds__(512) void fgl_fc_wmma_kernel(
    const float* __restrict__ agg,
    const float* __restrict__ v,      // [NOUT*COUT] (CIN==1)
    const float* __restrict__ g,      // [NOUT*COUT]
    const float* __restrict__ b_fgl,  // [NOUT*COUT]
    const float* __restrict__ W_fc,   // [NCLS, 16]
    const float* __restrict__ b_fc,   // [NCLS]
    float* __restrict__ out) {        // [NB, NCLS]
  const int tid     = threadIdx.x;
  const int lane    = tid & 31;
  const int wave    = tid >> 5;        // 0..15 -> M tile index
  const int rowBase = wave * 16;
  const int l16     = lane & 15;
  const bool hi     = lane >= 16;
  const int mrow    = rowBase + l16;   // A-matrix row owned by this lane
  const int ncol    = l16;             // B/C/D column owned by this lane

  // Padding mask for columns 10..15 (clamped index + multiplicative mask:
  // no exec-divergent loads).
  const float cmask = (ncol < NCLS) ? 1.0f : 0.0f;
  const int   ncc   = (ncol < NCLS) ? ncol : 0;

  // C accumulator seeded with b_fc (zero for padded columns)
  const float bias = cmask * b_fc[ncc];
  v8f c;
#pragma unroll
  for (int i = 0; i < 8; ++i) c[i] = bias;

#pragma unroll
  for (int kc = 0; kc < 4; ++kc) {
    const int k0 = kc * 4 + (hi ? 2 : 0);
    const int k1 = k0 + 1;

    // A fragment: y[mrow][k] = agg[mrow][k/4] * (g*sign(v))[k] + b_fgl[k]
    v2f a, b;
    const float aq = agg[mrow * NOUT + kc];
    a[0] = aq * __builtin_copysignf(g[k0], v[k0]) + b_fgl[k0];
    a[1] = aq * __builtin_copysignf(g[k1], v[k1]) + b_fgl[k1];

    // B fragment: B[k][n] = W_fc[n][k] for n<10, else 0 (padded columns)
    b[0] = cmask * W_fc[ncc * 16 + k0];
    b[1] = cmask * W_fc[ncc * 16 + k1];

    // D = A x B + C  (v_wmma_f32_16x16x4_f32)
    c = __builtin_amdgcn_wmma_f32_16x16x4_f32(
        /*neg_a=*/false, a, /*neg_b=*/false, b,
        /*c_mod=*/(short)0, c, /*reuse_a=*/false, /*reuse_b=*/false);
  }

  // D layout: vgpr i, lanes 0-15 -> row i; lanes 16-31 -> row i+8
#pragma unroll
  for (int i = 0; i < 8; ++i) {
    const int r = rowBase + i + (hi ? 8 : 0);
    if (ncol < NCLS) out[r * NCLS + ncol] = c[i];
  }
}

// ---------------------------------------------------------------------------
extern "C" void kernel_launch(void* const* d_in, const int* in_sizes, int n_in,
                              void* d_out, int out_size, void* d_ws, size_t ws_size,
                              hipStream_t stream) {
  (void)in_sizes; (void)n_in; (void)out_size; (void)ws_size;
  const float* x     = (const float*)d_in[0];
  const float* v     = (const float*)d_in[1];
  const float* g     = (const float*)d_in[2];
  const float* b_fgl = (const float*)d_in[3];
  const float* W_fc  = (const float*)d_in[4];
  const float* b_fc  = (const float*)d_in[5];
  float* out = (float*)d_out;
  float* agg = (float*)d_ws;  // [NB, NOUT] = 4 KiB scratch, fully overwritten

  dim3 grid(4, NB);  // 1024 blocks x 16 waves: saturate HBM
  quad_sum_kernel<<<grid, 512, 0, stream>>>(x, agg);
  fgl_fc_wmma_kernel<<<1, 512, 0, stream>>>(agg, v, g, b_fgl, W_fc, b_fc, out);
}